// AdaptiveEmbedding_22531398435527
// MI455X (gfx1250) — compile-verified
//
#include <hip/hip_runtime.h>

typedef __attribute__((ext_vector_type(2))) float v2f;
typedef __attribute__((ext_vector_type(8))) float v8f;

#define D_PROJ 1024
#define KC 128           // K-chunk staged in LDS
#define A_STRIDE 132     // 128 + 4 pad -> conflict-free ds reads (132 % 64 == 4)
#define M_ROWS 32        // two 16-token tiles per block (B-fragment reuse)

__global__ __launch_bounds__(256) void adaptive_emb_wmma_f32(
    const int* __restrict__ inp,
    const float* __restrict__ emb0, const float* __restrict__ proj0,
    const float* __restrict__ emb1, const float* __restrict__ proj1,
    const float* __restrict__ emb2, const float* __restrict__ proj2,
    const float* __restrict__ emb3, const float* __restrict__ proj3,
    float* __restrict__ out)
{
    __shared__ float A[M_ROWS * A_STRIDE];   // 32 tokens x K-chunk
    __shared__ int   srow[M_ROWS];           // in-bucket row index per token
    __shared__ int   sbuck[M_ROWS];          // bucket id per token
    __shared__ int   scnt[4];                // tokens per bucket in this tile

    const int tid      = threadIdx.x;
    const int lane     = tid & 31;       // wave32
    const int wid      = tid >> 5;       // 8 waves per block
    const int tok_base = blockIdx.x * M_ROWS;
    const int col_base = blockIdx.y * 128 + wid * 16;

    if (tid < 4) scnt[tid] = 0;
    __syncthreads();
    if (tid < M_ROWS) {
        const int id = inp[tok_base + tid];
        const int b  = (id < 20000) ? 0 : (id < 40000) ? 1 : (id < 200000) ? 2 : 3;
        const int start = (b == 0) ? 0 : (b == 1) ? 20000 : (b == 2) ? 40000 : 200000;
        srow[tid]  = id - start;
        sbuck[tid] = b;
        atomicAdd(&scnt[b], 1);
    }
    __syncthreads();

    const float* embs [4] = {emb0, emb1, emb2, emb3};
    const float* projs[4] = {proj0, proj1, proj2, proj3};
    const int    dims [4] = {1024, 256, 64, 16};

    v8f acc0 = {};   // tokens [tok_base,      +16)  x cols [col_base, +16)
    v8f acc1 = {};   // tokens [tok_base + 16, +16)  x cols [col_base, +16)

    // WMMA fragment coordinates (fixed per lane)
    const int frag_mn = lane & 15;            // M for A-frag, N for B-frag
    const int frag_kh = (lane >> 4) << 1;     // K offset 0 or 2 within the K=4 step

    for (int b = 0; b < 4; ++b) {
        if (scnt[b] == 0) continue;           // block-uniform skip of empty buckets
        const float* __restrict__ emb  = embs[b];
        const float* __restrict__ proj = projs[b];
        const int d = dims[b];

        for (int kb = 0; kb < d; kb += KC) {
            const int kc    = (d - kb) < KC ? (d - kb) : KC;   // 128, 64 or 16
            const int qper  = kc >> 2;                         // float4s per row (pow2)
            const int qsh   = __builtin_ctz(qper);
            const int total = M_ROWS << qsh;

            // Cooperative gather of 32 embedding-row slices into LDS (zeros if
            // the token is not in this bucket -> exact masked-add semantics).
            for (int f = tid; f < total; f += 256) {
                const int row = f >> qsh;
                const int k   = (f - (row << qsh)) << 2;
                float4 v = make_float4(0.f, 0.f, 0.f, 0.f);
                if (sbuck[row] == b) {
                    const float* src = emb + (size_t)srow[row] * d + kb + k;
                    v = *(const float4*)src;
                }
                *(float4*)(&A[row * A_STRIDE + k]) = v;
            }
            __syncthreads();

            // Two 16x16 output tiles per wave sharing each B fragment;
            // independent accumulators break the WMMA RAW chain.
            for (int kk = 0; kk < kc; kk += 4) {
                v2f bfrag = *(const v2f*)(&proj[(size_t)(col_base + frag_mn) * d
                                                + kb + kk + frag_kh]);
                v2f a0 = *(const v2f*)(&A[frag_mn * A_STRIDE + kk + frag_kh]);
                v2f a1 = *(const v2f*)(&A[(frag_mn + 16) * A_STRIDE + kk + frag_kh]);
                acc0 = __builtin_amdgcn_wmma_f32_16x16x4_f32(
                    false, a0, false, bfrag, (short)0, acc0, false, false);
                acc1 = __builtin_amdgcn_wmma_f32_16x16x4_f32(
                    false, a1, false, bfrag, (short)0, acc1, false, false);
            }
            __syncthreads();
        }
    }

    // C/D layout: VGPR r, lane l -> M = r + 8*(l>>4), N = l & 15
    const int n_out = lane & 15;
    const int m_hi  = (lane >> 4) << 3;
#pragma unroll
    for (int r = 0; r < 8; ++r) {
        out[(size_t)(tok_base + r + m_hi) * D_PROJ + col_base + n_out] = acc0[r];
    }
#pragma unroll
    for (int r = 0; r < 8; ++r) {
        out[(size_t)(tok_base + 16 + r + m_hi) * D_PROJ + col_base + n_out] = acc1[r];
    }
}

extern "C" void kernel_launch(void* const* d_in, const int* in_sizes, int n_in,
                              void* d_out, int out_size, void* d_ws, size_t ws_size,
                              hipStream_t stream) {
    // setup_inputs() dict order: inp, emb0, proj0, emb1, proj1, emb2, proj2, emb3, proj3
    const int*   inp   = (const int*)d_in[0];
    const float* emb0  = (const float*)d_in[1];
    const float* proj0 = (const float*)d_in[2];
    const float* emb1  = (const float*)d_in[3];
    const float* proj1 = (const float*)d_in[4];
    const float* emb2  = (const float*)d_in[5];
    const float* proj2 = (const float*)d_in[6];
    const float* emb3  = (const float*)d_in[7];
    const float* proj3 = (const float*)d_in[8];

    const int n_tok = in_sizes[0];                 // 16384
    dim3 grid((unsigned)(n_tok / M_ROWS), D_PROJ / 128);
    adaptive_emb_wmma_f32<<<grid, 256, 0, stream>>>(
        inp, emb0, proj0, emb1, proj1, emb2, proj2, emb3, proj3, (float*)d_out);
}